// CTCLossWithLabelSmoothing_46703474377104
// MI455X (gfx1250) — compile-verified
//
#include <hip/hip_runtime.h>
#include <hip/hip_bf16.h>

// Problem constants (match reference)
constexpr int T = 800;
constexpr int B = 32;
constexpr int V = 1024;
constexpr int S = 100;
constexpr int L = 2 * S + 1;        // 201 extended states
constexpr float NEG = -1e10f;
constexpr float LS = 0.1f;

constexpr int RED_WGS = 512;        // reduction workgroups
constexpr int RED_THREADS = 256;    // 8 wave32 per WG
// float4 count = T*B*V/4 = 6,553,600 ; threads = 131,072 -> exactly 50 f4/thread
constexpr int RED_ITERS = 25;       // 2 float4 per iteration (two streams)

typedef float v2f __attribute__((ext_vector_type(2)));
typedef float v8f __attribute__((ext_vector_type(8)));

// -------------------------------------------------------------------------
// Kernel 1: sum(log_probs) via V_WMMA_F32_16X16X4_F32 (exact f32 FMA).
// A = data tile (16x4 f32, 2 VGPRs/lane), B = ones(4x16) -> D rows = row sums.
// Two independent accumulator chains so both global_load_b128 per iteration
// are in flight before the first wait; compile-time trip count keeps loop
// control in SALU and EXEC all-1s (WMMA requirement).
// 16 duplicate D columns corrected by /16. One deterministic partial per WG.
// -------------------------------------------------------------------------
__global__ void __launch_bounds__(RED_THREADS)
smooth_reduce_kernel(const float* __restrict__ lp, float* __restrict__ partials) {
    const size_t stride = (size_t)RED_WGS * RED_THREADS;
    const size_t base = (size_t)blockIdx.x * RED_THREADS + threadIdx.x;

    v8f c0 = {0.f, 0.f, 0.f, 0.f, 0.f, 0.f, 0.f, 0.f};
    v8f c1 = {0.f, 0.f, 0.f, 0.f, 0.f, 0.f, 0.f, 0.f};
    v2f ones = {1.0f, 1.0f};
    const float4* __restrict__ x4 = (const float4*)lp;

    for (int k = 0; k < RED_ITERS; ++k) {
        float4 u = x4[base + (size_t)(2 * k)     * stride];
        float4 w = x4[base + (size_t)(2 * k + 1) * stride];
        v2f a0 = {u.x, u.y};
        v2f a1 = {u.z, u.w};
        v2f b0 = {w.x, w.y};
        v2f b1 = {w.z, w.w};
        // 8-arg form: (neg_a, A, neg_b, B, c_mod, C, reuse_a, reuse_b)
        c0 = __builtin_amdgcn_wmma_f32_16x16x4_f32(false, a0, false, ones,
                                                   (short)0, c0, false, false);
        c0 = __builtin_amdgcn_wmma_f32_16x16x4_f32(false, a1, false, ones,
                                                   (short)0, c0, false, false);
        c1 = __builtin_amdgcn_wmma_f32_16x16x4_f32(false, b0, false, ones,
                                                   (short)0, c1, false, false);
        c1 = __builtin_amdgcn_wmma_f32_16x16x4_f32(false, b1, false, ones,
                                                   (short)0, c1, false, false);
    }

    float s = (c0[0] + c0[1] + c0[2] + c0[3] + c0[4] + c0[5] + c0[6] + c0[7])
            + (c1[0] + c1[1] + c1[2] + c1[3] + c1[4] + c1[5] + c1[6] + c1[7]);

    __shared__ float red[RED_THREADS];
    red[threadIdx.x] = s;
    __syncthreads();
    for (int off = RED_THREADS / 2; off > 0; off >>= 1) {
        if ((int)threadIdx.x < off) red[threadIdx.x] += red[threadIdx.x + off];
        __syncthreads();
    }
    if (threadIdx.x == 0)
        partials[blockIdx.x] = red[0] * (1.0f / 16.0f);   // 16 duplicate columns
}

// -------------------------------------------------------------------------
// Kernel 2: CTC alpha recursion. One workgroup per batch element.
// Thread s holds alpha[s] in a register; neighbors exchanged through LDS.
// Emission gathers are prefetched one time-step ahead to hide global latency.
// -------------------------------------------------------------------------
__device__ __forceinline__ float logaddexp_f(float a, float b) {
    float m = fmaxf(a, b);
    return m + log1pf(expf(-fabsf(a - b)));
}

__global__ void __launch_bounds__(256)
ctc_alpha_kernel(const float* __restrict__ lp,
                 const int* __restrict__ targets,
                 const int* __restrict__ input_len,
                 const int* __restrict__ target_len,
                 float* __restrict__ nll_out) {
    const int b = blockIdx.x;
    const int s = threadIdx.x;
    const bool active = (s < L);

    __shared__ float sh[2 + 256];     // sh[0..1] = NEG pads for s-1 / s-2
    float* shp = sh + 2;
    if (threadIdx.x < 2) sh[threadIdx.x] = NEG;

    // extended label for state s: even -> blank(0), odd -> targets[b][(s-1)/2]
    int lab = 0;
    if (active && (s & 1)) lab = targets[b * S + (s >> 1)];
    // can_skip: odd s>=3 and label differs from label two back
    bool can_skip = false;
    if (active && (s & 1) && s >= 3)
        can_skip = (lab != targets[b * S + (s >> 1) - 1]);

    const int len = input_len[b];                       // in [T/2, T]
    const float* lpb = lp + (size_t)b * V;              // row t at lpb + t*B*V
    const size_t rowstride = (size_t)B * V;

    // t = 0 init
    float alpha = NEG;
    if (s < 2) alpha = lpb[lab];                        // s==0 -> lab==0 (blank)

    // prefetch emission for t = 1
    float e_next = lpb[rowstride + (size_t)lab];

    for (int t = 1; t < len; ++t) {
        float e = e_next;
        int tn = (t + 1 < len) ? (t + 1) : t;           // clamp: stay in-bounds
        e_next = lpb[(size_t)tn * rowstride + (size_t)lab];

        shp[s] = alpha;
        __syncthreads();
        float a1 = shp[s - 1];
        float a2 = shp[s - 2];
        float nw = logaddexp_f(alpha, a1);
        if (can_skip) nw = logaddexp_f(nw, a2);
        alpha = active ? (nw + e) : NEG;
        __syncthreads();
    }

    // final: nll = -logaddexp(alpha[2*tl], alpha[2*tl-1]); zero_infinity; /tl
    shp[s] = alpha;
    __syncthreads();
    if (threadIdx.x == 0) {
        int tl = target_len[b];
        int idx = 2 * tl;                               // <= 200
        float ab = shp[idx];
        float al = shp[idx - 1];
        float nll = -logaddexp_f(ab, al);
        if (nll > 1e9f) nll = 0.0f;                     // zero_infinity
        nll_out[b] = nll / (float)tl;
    }
}

// -------------------------------------------------------------------------
// Kernel 3: deterministic finalize.
// out = (1-LS) * mean(nll/tl) + LS * (-sum(lp) / (T*B*V))
// -------------------------------------------------------------------------
__global__ void __launch_bounds__(256)
finalize_kernel(const float* __restrict__ partials,   // RED_WGS entries
                const float* __restrict__ nll,        // B entries
                float* __restrict__ out) {
    __shared__ float red[256];
    // reduce smooth partials (512 -> 1), fixed tree order
    float s = partials[threadIdx.x] + partials[threadIdx.x + 256];
    red[threadIdx.x] = s;
    __syncthreads();
    for (int off = 128; off > 0; off >>= 1) {
        if ((int)threadIdx.x < off) red[threadIdx.x] += red[threadIdx.x + off];
        __syncthreads();
    }
    float smooth_sum = red[0];
    __syncthreads();

    // reduce nll (32 -> 1)
    red[threadIdx.x] = ((int)threadIdx.x < B) ? nll[threadIdx.x] : 0.0f;
    __syncthreads();
    for (int off = 16; off > 0; off >>= 1) {
        if ((int)threadIdx.x < off) red[threadIdx.x] += red[threadIdx.x + off];
        __syncthreads();
    }
    if (threadIdx.x == 0) {
        float ctc = red[0] / (float)B;
        float smooth = -smooth_sum / (float)((size_t)T * B * V);
        out[0] = (1.0f - LS) * ctc + LS * smooth;
    }
}

extern "C" void kernel_launch(void* const* d_in, const int* in_sizes, int n_in,
                              void* d_out, int out_size, void* d_ws, size_t ws_size,
                              hipStream_t stream) {
    const float* lp  = (const float*)d_in[0];   // (T,B,V) f32
    const int* tgt   = (const int*)d_in[1];     // (B,S)  i32
    const int* ilen  = (const int*)d_in[2];     // (B,)   i32
    const int* tlen  = (const int*)d_in[3];     // (B,)   i32
    float* out = (float*)d_out;

    float* partials = (float*)d_ws;             // RED_WGS floats
    float* nll      = partials + RED_WGS;       // B floats

    smooth_reduce_kernel<<<RED_WGS, RED_THREADS, 0, stream>>>(lp, partials);
    ctc_alpha_kernel<<<B, 256, 0, stream>>>(lp, tgt, ilen, tlen, nll);
    finalize_kernel<<<1, 256, 0, stream>>>(partials, nll, out);
}